// ModelSglang_28681791602762
// MI455X (gfx1250) — compile-verified
//
#include <hip/hip_runtime.h>

// Paged-KV index build for MI455X (gfx1250, wave32).
//
// out[b,p] = req_to_token[req_pool_indices[b], p*64] >> 6  for p < ceil(len[b]/64), else 0
//
// Memory-latency-bound strided gather (each gathered element sits on its own
// HBM cacheline). Strategy:
//  - one block per request  -> metadata becomes scalar loads (s_load_b32)
//  - branch-free masking via index clamp -> 4 global_load_b32 NT in flight
//    per thread (load clause), one wait, then shift+cndmask
//  - masked lanes all load the same clamped line -> ~2x traffic saving kept
//  - 4 contiguous pages per thread -> single b128 NT streaming store
// No matmul structure -> WMMA not applicable.

#define PAGE_SHIFT 6
#define ROW_SHIFT  16      // log2(MAX_CTX = 65536), row stride of req_to_token
#define TPB 256

// Native clang vector type: required by __builtin_nontemporal_store
// (HIP's int4 is a struct and is rejected).
typedef __attribute__((ext_vector_type(4))) int v4i;

__global__ __launch_bounds__(TPB) void kv_page_index_kernel(
    const int* __restrict__ req_to_token,      // [POOL, 65536]
    const int* __restrict__ req_pool_indices,  // [B]
    const int* __restrict__ page_kernel_lens,  // [B]
    int* __restrict__ out)                     // [B, 1024]
{
    const int b = blockIdx.x;

    // Uniform per-block metadata -> scalar loads.
    const int row = req_pool_indices[b];
    const int len = page_kernel_lens[b];
    const int np  = (len + ((1 << PAGE_SHIFT) - 1)) >> PAGE_SHIFT;
    const int last = (np > 0) ? (np - 1) : 0;   // clamped gather index (scalar)

    const int* __restrict__ rp = req_to_token + ((size_t)row << ROW_SHIFT);

    // 4 contiguous pages per thread: p0 .. p0+3
    const int p0 = (int)threadIdx.x << 2;

    // Phase 1: branch-free clamped gathers. No load result is consumed here,
    // so all 4 NT loads issue back-to-back before any s_wait_loadcnt.
    int raw[4];
#pragma unroll
    for (int i = 0; i < 4; ++i) {
        const int p  = p0 + i;
        const int pc = (p < np) ? p : last;     // in-bounds; masked lanes share a line
        raw[i] = __builtin_nontemporal_load(rp + ((size_t)pc << PAGE_SHIFT));
    }

    // Phase 2: shift + mask (tok in [0,2^28) -> shift == floor div), then one
    // 16B NT streaming store per thread (masked pages written as 0 — harness
    // poisons d_out, every element must be written).
    v4i o;
    o.x = (p0 + 0 < np) ? (raw[0] >> PAGE_SHIFT) : 0;
    o.y = (p0 + 1 < np) ? (raw[1] >> PAGE_SHIFT) : 0;
    o.z = (p0 + 2 < np) ? (raw[2] >> PAGE_SHIFT) : 0;
    o.w = (p0 + 3 < np) ? (raw[3] >> PAGE_SHIFT) : 0;

    v4i* __restrict__ orow = (v4i*)(out + ((size_t)b << 10));
    __builtin_nontemporal_store(o, orow + threadIdx.x);
}

// Generic fallback (any P): one thread per page, same clamp trick.
__global__ void kv_page_index_generic(
    const int* __restrict__ req_to_token,
    const int* __restrict__ req_pool_indices,
    const int* __restrict__ page_kernel_lens,
    int* __restrict__ out, int P)
{
    const int b = blockIdx.y;
    const int p = blockIdx.x * blockDim.x + threadIdx.x;
    if (p >= P) return;
    const int row = req_pool_indices[b];
    const int len = page_kernel_lens[b];
    const int np  = (len + ((1 << PAGE_SHIFT) - 1)) >> PAGE_SHIFT;
    const int pc  = (p < np) ? p : ((np > 0) ? np - 1 : 0);
    const int tok = __builtin_nontemporal_load(
        req_to_token + ((size_t)row << ROW_SHIFT) + ((size_t)pc << PAGE_SHIFT));
    const int v = (p < np) ? (tok >> PAGE_SHIFT) : 0;
    __builtin_nontemporal_store(v, out + (size_t)b * P + p);
}

extern "C" void kernel_launch(void* const* d_in, const int* in_sizes, int n_in,
                              void* d_out, int out_size, void* d_ws, size_t ws_size,
                              hipStream_t stream) {
    const int* req_to_token     = (const int*)d_in[0];
    const int* req_pool_indices = (const int*)d_in[1];
    const int* page_kernel_lens = (const int*)d_in[2];
    int* out = (int*)d_out;

    const int B = in_sizes[1];            // 1024 requests
    const int P = out_size / B;           // 1024 pages per request

    if (P == 1024) {
        // Specialized branch-free path: 256 threads x 4 contiguous pages.
        kv_page_index_kernel<<<dim3(B), dim3(TPB), 0, stream>>>(
            req_to_token, req_pool_indices, page_kernel_lens, out);
    } else {
        dim3 grid((P + TPB - 1) / TPB, B);
        kv_page_index_generic<<<grid, dim3(TPB), 0, stream>>>(
            req_to_token, req_pool_indices, page_kernel_lens, out, P);
    }
}